// SparseAttLayer_3607772529175
// MI455X (gfx1250) — compile-verified
//
#include <hip/hip_runtime.h>

typedef __attribute__((ext_vector_type(2))) float v2f;
typedef __attribute__((ext_vector_type(8))) float v8f;

#define B_TOT 8192
#define F_DIM 128
#define E_DIM 64
#define O_DIM 64

#define X_STRIDE  68   // floats per X LDS row; 16B-aligned rows, conflict-free A fetch
#define G_STRIDE  65   // floats per G LDS row; conflict-free f-strided entmax reads
#define Q2_STRIDE 80   // float2 per kp-row of interleaved Q^T; K-halves land 16 pair-banks apart

__launch_bounds__(256)
__global__ void sparse_att_entmax_kernel(const float* __restrict__ x,
                                         const float* __restrict__ Q,
                                         const float* __restrict__ values,
                                         float* __restrict__ out)
{
    __shared__ float Xl[F_DIM * X_STRIDE];            // 34816 B; reused as G after GEMM
    __shared__ v2f   Qt2[(E_DIM / 2) * Q2_STRIDE];    // 20480 B; Qt2[kp][n] = {Q[n][2kp], Q[n][2kp+1]}

    const int tid  = threadIdx.x;
    const int lane = tid & 31;
    const int wave = tid >> 5;
    const int b    = blockIdx.x;

    // ---- stage X_b (128x64) via async global->LDS b128; Q^T interleaved via VGPRs ----
    {
        const float* xg = x + (size_t)b * (F_DIM * E_DIM);
        #pragma unroll
        for (int i = tid; i < (F_DIM * E_DIM / 4); i += 256) {
            const int f  = i >> 4;             // 16 float4 per 64-float row
            const int e4 = (i & 15) << 2;
            unsigned lds_off = (unsigned)(size_t)&Xl[f * X_STRIDE + e4]; // low 32b = LDS byte offset
            const float* g = xg + (i << 2);
            asm volatile("global_load_async_to_lds_b128 %0, %1, off"
                         :: "v"(lds_off), "v"(g) : "memory");
        }
        float* q2 = (float*)Qt2;
        #pragma unroll
        for (int i = tid; i < O_DIM * E_DIM; i += 256) {
            const int o = i >> 6;
            const int e = i & 63;
            q2[(e >> 1) * (Q2_STRIDE * 2) + (o << 1) + (e & 1)] = Q[i];
        }
        asm volatile("s_wait_asynccnt 0x0" ::: "memory");
    }
    __syncthreads();

    // ---- GEMM: G[f,o] = sum_e X[f,e]*Q[o,e] via V_WMMA_F32_16X16X4_F32 ----
    // wave w owns f-tile [16w,16w+16); N split into 4 tiles of 16 o.
    const int mlo    = lane & 15;     // M (or N) index within a 16-wide tile
    const int kp_off = lane >> 4;     // lanes 0-15 -> K{0,1}; lanes 16-31 -> K{2,3}

    v8f acc[4] = {};
    #pragma unroll
    for (int kk = 0; kk < 16; ++kk) {
        const int k  = (kk << 2) + (kp_off << 1);
        const int kp = (kk << 1) + kp_off;
        v2f a = *(const v2f*)&Xl[(wave * 16 + mlo) * X_STRIDE + k];   // one ds_load_b64
        #pragma unroll
        for (int nt = 0; nt < 4; ++nt) {
            v2f bm = Qt2[kp * Q2_STRIDE + nt * 16 + mlo];             // one ds_load_b64
            acc[nt] = __builtin_amdgcn_wmma_f32_16x16x4_f32(
                false, a, false, bm, (short)0, acc[nt], false, false);
        }
    }

    __syncthreads();                 // all waves done reading Xl -> reuse as G
    float* G = Xl;

    // C layout: VGPR r: lanes 0-15 -> M=r, lanes 16-31 -> M=r+8; N = lane&15
    {
        const int fbase = wave * 16 + ((lane >> 4) << 3);
        #pragma unroll
        for (int nt = 0; nt < 4; ++nt) {
            const int o = nt * 16 + mlo;
            #pragma unroll
            for (int r = 0; r < 8; ++r)
                G[(fbase + r) * G_STRIDE + o] = acc[nt][r];
        }
    }
    __syncthreads();

    // ---- entmax-1.5 bisection over f (d=128), 50 iters; wave owns 8 o-rows ----
    // Xs = G * (alpha-1) = G * 0.5 ; p(z) = relu(z)^2
    float xs[8][4];
    #pragma unroll
    for (int oo = 0; oo < 8; ++oo) {
        const int o = wave * 8 + oo;
        #pragma unroll
        for (int i = 0; i < 4; ++i)
            xs[oo][i] = G[(lane + 32 * i) * G_STRIDE + o] * 0.5f;
    }

    float tau_lo[8];
    #pragma unroll
    for (int oo = 0; oo < 8; ++oo) {
        float m = fmaxf(fmaxf(xs[oo][0], xs[oo][1]), fmaxf(xs[oo][2], xs[oo][3]));
        #pragma unroll
        for (int mask = 16; mask >= 1; mask >>= 1)
            m = fmaxf(m, __shfl_xor(m, mask, 32));
        tau_lo[oo] = m - 1.0f;       // gp(1)
    }

    float p[8][4];
    float s[8];
    float dm = 0.91161165235168155f; // (max - tau_lo) - (max - tau_hi) = 1 - 128^(-1/2)

    for (int it = 0; it < 50; ++it) {
        dm *= 0.5f;
        float tau[8];
        #pragma unroll
        for (int oo = 0; oo < 8; ++oo) {
            tau[oo] = tau_lo[oo] + dm;
            float a0 = 0.0f;
            #pragma unroll
            for (int i = 0; i < 4; ++i) {
                float t = fmaxf(xs[oo][i] - tau[oo], 0.0f);
                t = t * t;
                p[oo][i] = t;
                a0 += t;
            }
            s[oo] = a0;
        }
        #pragma unroll
        for (int mask = 16; mask >= 1; mask >>= 1) {
            #pragma unroll
            for (int oo = 0; oo < 8; ++oo)
                s[oo] += __shfl_xor(s[oo], mask, 32);
        }
        #pragma unroll
        for (int oo = 0; oo < 8; ++oo)
            if (s[oo] >= 1.0f) tau_lo[oo] = tau[oo];   // f_m = s-1 >= 0
    }

    // ---- renorm with last-iteration p,s; scale by values; coalesced stores ----
    float* outb = out + (size_t)b * (O_DIM * F_DIM);
    #pragma unroll
    for (int oo = 0; oo < 8; ++oo) {
        const int o = wave * 8 + oo;
        const float inv = 1.0f / s[oo];
        #pragma unroll
        for (int i = 0; i < 4; ++i) {
            const int f = lane + 32 * i;
            outb[o * F_DIM + f] = p[oo][i] * inv * values[o * F_DIM + f];
        }
    }
}

extern "C" void kernel_launch(void* const* d_in, const int* in_sizes, int n_in,
                              void* d_out, int out_size, void* d_ws, size_t ws_size,
                              hipStream_t stream) {
    (void)in_sizes; (void)n_in; (void)out_size; (void)d_ws; (void)ws_size;
    const float* x      = (const float*)d_in[0];
    const float* Q      = (const float*)d_in[1];
    const float* values = (const float*)d_in[2];
    float* out          = (float*)d_out;
    sparse_att_entmax_kernel<<<B_TOT, 256, 0, stream>>>(x, Q, values, out);
}